// GoEModel_74199855006277
// MI455X (gfx1250) — compile-verified
//
#include <hip/hip_runtime.h>
#include <math.h>

#define B_ 2048
#define S_ 128
#define D_ 512
#define H_ 2048
#define HR_ 512
#define E_ 8
#define C_ 16
#define PATH_ 5
#define NEG_ (-1e9f)

typedef float v2f __attribute__((ext_vector_type(2)));
typedef float v4f __attribute__((ext_vector_type(4)));
typedef float v8f __attribute__((ext_vector_type(8)));

__device__ __forceinline__ v8f wmma4(v2f a, v2f b, v8f c) {
  // D = A(16x4 f32) * B(4x16 f32) + C(16x16 f32)
  return __builtin_amdgcn_wmma_f32_16x16x4_f32(false, a, false, b, (short)0, c,
                                               false, false);
}

__device__ __forceinline__ v8f zero8() {
  v8f z;
#pragma unroll
  for (int q = 0; q < 8; ++q) z[q] = 0.f;
  return z;
}

// ---------------- init ----------------
__global__ void k_init(int* active, int* visits, float* total_ent) {
  int b = blockIdx.x * 256 + threadIdx.x;
  if (b < B_) {
    active[b] = 1;
#pragma unroll
    for (int e = 0; e < E_; ++e) visits[b * E_ + e] = 0;
  }
  if (b == 0) *total_ent = 0.f;
}

// ---------------- embedding mean pool ----------------
__global__ void k_embed(const int* __restrict__ ids, const float* __restrict__ emb,
                        float* __restrict__ rep) {
  __shared__ int sid[S_];
  const int b = blockIdx.x;
  const int t = threadIdx.x;  // 128 threads, one v4f column each
  sid[t] = ids[b * S_ + t];
  __syncthreads();
  v4f acc;
#pragma unroll
  for (int q = 0; q < 4; ++q) acc[q] = 0.f;
  for (int s = 0; s < S_; ++s) {
    const v4f* p = reinterpret_cast<const v4f*>(emb + (size_t)sid[s] * D_);
    acc += p[t];
  }
  acc *= (1.f / (float)S_);
  reinterpret_cast<v4f*>(rep + (size_t)b * D_)[t] = acc;
}

// ---------------- per-step scratch reset ----------------
__global__ void k_step_init(int* counts, float* sum_ent, int* n_act) {
  int t = threadIdx.x;
  if (t < E_) counts[t] = 0;
  if (t == 8) *sum_ent = 0.f;
  if (t == 9) *n_act = 0;
}

// ---------------- router: FC1(WMMA) + logits + softmax/entropy/argmax ----------------
__global__ void k_router(const float* __restrict__ rep,
                         const float* __restrict__ rW1, const float* __restrict__ rb1,
                         const float* __restrict__ rW2, const float* __restrict__ rb2,
                         const int* __restrict__ visits, const int* __restrict__ active,
                         int* __restrict__ decision, float* sum_ent, int* n_act) {
  __shared__ float sA[16 * 132];   // 16-row A chunk, padded stride
  __shared__ float sH[16 * 520];   // relu(hidden) tile [16][512]
  __shared__ float sL[16 * 16];    // masked logits [16][9]
  const int t = threadIdx.x;
  const int w = t >> 5;
  const int lane = t & 31;
  const int rb = blockIdx.x * 16;
  const int m = lane & 15;          // A row / B col within tile
  const int kh = (lane >> 4) * 2;   // K sub-offset per half-wave

  v8f acc[8];
#pragma unroll
  for (int j = 0; j < 8; ++j) acc[j] = zero8();

  for (int kc = 0; kc < 4; ++kc) {
    {  // cooperative A-chunk load: rows rb..rb+15, cols kc*128..+127
      int i = t >> 3, jj = t & 7;
      const float* src = rep + (size_t)(rb + i) * D_ + kc * 128 + jj * 16;
      float* dst = sA + i * 132 + jj * 16;
#pragma unroll
      for (int q = 0; q < 4; ++q)
        reinterpret_cast<v4f*>(dst)[q] = reinterpret_cast<const v4f*>(src)[q];
    }
    __syncthreads();
    for (int s = 0; s < 32; ++s) {
      int kk = s * 4 + kh;
      v2f a;
      a.x = sA[m * 132 + kk];
      a.y = sA[m * 132 + kk + 1];
      int kg = kc * 128 + kk;
#pragma unroll
      for (int j = 0; j < 8; ++j) {
        int n = (w + 4 * j) * 16 + m;
        v2f bb;
        bb.x = rW1[(size_t)kg * HR_ + n];
        bb.y = rW1[(size_t)(kg + 1) * HR_ + n];
        acc[j] = wmma4(a, bb, acc[j]);
      }
    }
    __syncthreads();
  }
  // bias + relu -> sH
#pragma unroll
  for (int j = 0; j < 8; ++j) {
    int n = (w + 4 * j) * 16 + m;
    float bias = rb1[n];
#pragma unroll
    for (int g = 0; g < 8; ++g) {
      int row = g + 8 * (lane >> 4);
      float v = acc[j][g] + bias;
      sH[row * 520 + n] = v > 0.f ? v : 0.f;
    }
  }
  __syncthreads();
  // logits [16][9] (tiny N: plain VALU dot products)
  if (t < 16 * (E_ + 1)) {
    int r = t / (E_ + 1), c = t % (E_ + 1);
    float sum = rb2[c];
    for (int k = 0; k < HR_; ++k) sum += sH[r * 520 + k] * rW2[k * (E_ + 1) + c];
    int b = rb + r;
    if (c < E_ && visits[b * E_ + c] >= 1) sum = NEG_;  // MAX_VISITS = 1
    sL[r * 16 + c] = sum;
  }
  __syncthreads();
  // per-row softmax entropy + argmax decision
  if (t < 16) {
    int r = t, b = rb + r;
    float mx = sL[r * 16];
    int arg = 0;
#pragma unroll
    for (int c = 1; c < E_ + 1; ++c) {
      float v = sL[r * 16 + c];
      if (v > mx) { mx = v; arg = c; }
    }
    float p[E_ + 1];
    float Z = 0.f;
#pragma unroll
    for (int c = 0; c < E_ + 1; ++c) {
      p[c] = __expf(sL[r * 16 + c] - mx);
      Z += p[c];
    }
    float inv = 1.f / Z, ent = 0.f;
#pragma unroll
    for (int c = 0; c < E_ + 1; ++c) {
      float pp = p[c] * inv;
      ent -= pp * __logf(pp + 1e-9f);
    }
    decision[b] = arg;
    if (active[b]) {
      atomicAdd(sum_ent, ent);
      atomicAdd(n_act, 1);
    }
  }
}

// ---------------- entropy finalize + counting-sort by expert ----------------
__global__ void k_partition(int* active, const int* __restrict__ decision,
                            int* __restrict__ visits, int* counts, int* lists,
                            const float* sum_ent, const int* n_act,
                            float* total_ent) {
  int b = blockIdx.x * 128 + threadIdx.x;
  if (b == 0) {
    int na = *n_act;
    if (na > 0) *total_ent += *sum_ent / (float)na;
  }
  if (b < B_) {
    if (active[b]) {
      int dec = decision[b];
      if (dec == E_) {
        active[b] = 0;  // terminate; rep stays frozen == final
      } else {
        int slot = atomicAdd(&counts[dec], 1);
        lists[dec * B_ + slot] = b;
        visits[b * E_ + dec] += 1;
      }
    }
  }
}

// ---------------- grouped expert FC1: hh = relu(rep @ eW1[e] + eb1[e]) ----------------
__global__ void k_fc1(const float* __restrict__ rep, const float* __restrict__ eW1,
                      const float* __restrict__ eb1, float* __restrict__ hh,
                      const int* __restrict__ counts, const int* __restrict__ lists) {
  const int e = blockIdx.z;
  const int cnt = counts[e];
  const int row0 = blockIdx.x * 16;
  if (row0 >= cnt) return;
  __shared__ int sRow[16];
  __shared__ int sVal[16];
  __shared__ float sA[16 * 132];
  const int t = threadIdx.x, w = t >> 5, lane = t & 31;
  if (t < 16) {
    int ok = (row0 + t) < cnt;
    sVal[t] = ok;
    sRow[t] = lists[e * B_ + (ok ? row0 + t : row0)];
  }
  __syncthreads();
  const int m = lane & 15;
  const int kh = (lane >> 4) * 2;
  const int nb = blockIdx.y * 64 + w * 16;
  const float* wp = eW1 + (size_t)e * D_ * H_;
  v8f acc = zero8();
  for (int kc = 0; kc < 4; ++kc) {
    {
      int i = t >> 3, jj = t & 7;
      const float* src = rep + (size_t)sRow[i] * D_ + kc * 128 + jj * 16;
      float* dst = sA + i * 132 + jj * 16;
#pragma unroll
      for (int q = 0; q < 4; ++q)
        reinterpret_cast<v4f*>(dst)[q] = reinterpret_cast<const v4f*>(src)[q];
    }
    __syncthreads();
    for (int s = 0; s < 32; ++s) {
      int kk = s * 4 + kh;
      v2f a;
      a.x = sA[m * 132 + kk];
      a.y = sA[m * 132 + kk + 1];
      int kg = kc * 128 + kk;
      v2f bb;
      bb.x = wp[(size_t)kg * H_ + nb + m];
      bb.y = wp[(size_t)(kg + 1) * H_ + nb + m];
      acc = wmma4(a, bb, acc);
    }
    __syncthreads();
  }
  const int n = nb + m;
  const float bias = eb1[e * H_ + n];
#pragma unroll
  for (int g = 0; g < 8; ++g) {
    int row = g + 8 * (lane >> 4);
    if (sVal[row]) {
      float v = acc[g] + bias;
      hh[(size_t)sRow[row] * H_ + n] = v > 0.f ? v : 0.f;
    }
  }
}

// ---------------- grouped expert FC2: rep = hh @ eW2[e] + eb2[e] + etag[e] ----------------
__global__ void k_fc2(const float* __restrict__ hh, const float* __restrict__ eW2,
                      const float* __restrict__ eb2, const float* __restrict__ etag,
                      float* __restrict__ rep, const int* __restrict__ counts,
                      const int* __restrict__ lists) {
  const int e = blockIdx.z;
  const int cnt = counts[e];
  const int row0 = blockIdx.x * 16;
  if (row0 >= cnt) return;
  __shared__ int sRow[16];
  __shared__ int sVal[16];
  __shared__ float sA[16 * 132];
  const int t = threadIdx.x, w = t >> 5, lane = t & 31;
  if (t < 16) {
    int ok = (row0 + t) < cnt;
    sVal[t] = ok;
    sRow[t] = lists[e * B_ + (ok ? row0 + t : row0)];
  }
  __syncthreads();
  const int m = lane & 15;
  const int kh = (lane >> 4) * 2;
  const int nb = blockIdx.y * 64 + w * 16;
  const float* wp = eW2 + (size_t)e * H_ * D_;
  v8f acc = zero8();
  for (int kc = 0; kc < 16; ++kc) {
    {
      int i = t >> 3, jj = t & 7;
      const float* src = hh + (size_t)sRow[i] * H_ + kc * 128 + jj * 16;
      float* dst = sA + i * 132 + jj * 16;
#pragma unroll
      for (int q = 0; q < 4; ++q)
        reinterpret_cast<v4f*>(dst)[q] = reinterpret_cast<const v4f*>(src)[q];
    }
    __syncthreads();
    for (int s = 0; s < 32; ++s) {
      int kk = s * 4 + kh;
      v2f a;
      a.x = sA[m * 132 + kk];
      a.y = sA[m * 132 + kk + 1];
      int kg = kc * 128 + kk;
      v2f bb;
      bb.x = wp[(size_t)kg * D_ + nb + m];
      bb.y = wp[(size_t)(kg + 1) * D_ + nb + m];
      acc = wmma4(a, bb, acc);
    }
    __syncthreads();
  }
  const int n = nb + m;
  const float bias = eb2[e * D_ + n] + etag[e * D_ + n];
#pragma unroll
  for (int g = 0; g < 8; ++g) {
    int row = g + 8 * (lane >> 4);
    if (sVal[row]) rep[(size_t)sRow[row] * D_ + n] = acc[g] + bias;
  }
}

// ---------------- classifier: out = rep @ cW + cb ; out[B*C] = total_ent ----------------
__global__ void k_cls(const float* __restrict__ rep, const float* __restrict__ cW,
                      const float* __restrict__ cb, const float* total_ent,
                      float* __restrict__ out) {
  __shared__ float sA[16 * 132];
  const int t = threadIdx.x, w = t >> 5, lane = t & 31;
  const int rb = blockIdx.x * 16;
  const int m = lane & 15;
  const int kh = (lane >> 4) * 2;
  v8f acc = zero8();
  for (int kc = 0; kc < 4; ++kc) {
    {
      int i = t >> 3, jj = t & 7;
      const float* src = rep + (size_t)(rb + i) * D_ + kc * 128 + jj * 16;
      float* dst = sA + i * 132 + jj * 16;
#pragma unroll
      for (int q = 0; q < 4; ++q)
        reinterpret_cast<v4f*>(dst)[q] = reinterpret_cast<const v4f*>(src)[q];
    }
    __syncthreads();
    if (w == 0) {  // single wave computes (EXEC all-ones within wave)
      for (int s = 0; s < 32; ++s) {
        int kk = s * 4 + kh;
        v2f a;
        a.x = sA[m * 132 + kk];
        a.y = sA[m * 132 + kk + 1];
        int kg = kc * 128 + kk;
        v2f bb;
        bb.x = cW[(size_t)kg * C_ + m];
        bb.y = cW[(size_t)(kg + 1) * C_ + m];
        acc = wmma4(a, bb, acc);
      }
    }
    __syncthreads();
  }
  if (w == 0) {
    const float bias = cb[m];
#pragma unroll
    for (int g = 0; g < 8; ++g) {
      int row = g + 8 * (lane >> 4);
      out[(size_t)(rb + row) * C_ + m] = acc[g] + bias;
    }
  }
  if (blockIdx.x == 0 && t == 0) out[B_ * C_] = *total_ent;
}

extern "C" void kernel_launch(void* const* d_in, const int* in_sizes, int n_in,
                              void* d_out, int out_size, void* d_ws, size_t ws_size,
                              hipStream_t stream) {
  (void)in_sizes; (void)n_in; (void)out_size; (void)ws_size;
  const int*   ids  = (const int*)d_in[0];
  const float* emb  = (const float*)d_in[1];
  const float* rW1  = (const float*)d_in[2];
  const float* rb1  = (const float*)d_in[3];
  const float* rW2  = (const float*)d_in[4];
  const float* rb2  = (const float*)d_in[5];
  const float* eW1  = (const float*)d_in[6];
  const float* eb1  = (const float*)d_in[7];
  const float* eW2  = (const float*)d_in[8];
  const float* eb2  = (const float*)d_in[9];
  const float* etag = (const float*)d_in[10];
  const float* cW   = (const float*)d_in[11];
  const float* cb   = (const float*)d_in[12];
  float* out = (float*)d_out;

  // workspace layout
  float* rep      = (float*)d_ws;                       // B*D
  float* hh       = rep + (size_t)B_ * D_;              // B*H
  int*   decision = (int*)(hh + (size_t)B_ * H_);       // B
  int*   active   = decision + B_;                      // B
  int*   visits   = active + B_;                        // B*E
  int*   counts   = visits + B_ * E_;                   // E
  int*   lists    = counts + E_;                        // E*B
  float* sum_ent  = (float*)(lists + E_ * B_);          // 1
  int*   n_act    = (int*)(sum_ent + 1);                // 1
  float* total_ent = (float*)(n_act + 1);               // 1

  k_init<<<(B_ + 255) / 256, 256, 0, stream>>>(active, visits, total_ent);
  k_embed<<<B_, 128, 0, stream>>>(ids, emb, rep);
  for (int step = 0; step < PATH_; ++step) {
    k_step_init<<<1, 32, 0, stream>>>(counts, sum_ent, n_act);
    k_router<<<B_ / 16, 128, 0, stream>>>(rep, rW1, rb1, rW2, rb2, visits,
                                          active, decision, sum_ent, n_act);
    k_partition<<<B_ / 128, 128, 0, stream>>>(active, decision, visits, counts,
                                              lists, sum_ent, n_act, total_ent);
    k_fc1<<<dim3(B_ / 16, H_ / 64, E_), 128, 0, stream>>>(rep, eW1, eb1, hh,
                                                          counts, lists);
    k_fc2<<<dim3(B_ / 16, D_ / 64, E_), 128, 0, stream>>>(hh, eW2, eb2, etag,
                                                          rep, counts, lists);
  }
  k_cls<<<B_ / 16, 128, 0, stream>>>(rep, cW, cb, total_ent, out);
}